// Quantize_37512244363905
// MI455X (gfx1250) — compile-verified
//
#include <hip/hip_runtime.h>

typedef __attribute__((ext_vector_type(2))) float v2f;
typedef __attribute__((ext_vector_type(4))) float v4f;
typedef __attribute__((ext_vector_type(8))) float v8f;
typedef __attribute__((ext_vector_type(4))) unsigned int v4u;
typedef __attribute__((ext_vector_type(8))) int v8i;
typedef __attribute__((ext_vector_type(4))) int v4i;

static constexpr int kDim    = 256;
static constexpr int kNEmbed = 512;
static constexpr int kB      = 16;
static constexpr int kT      = 2048;
static constexpr float kInvTotal = 1.0f / (16.0f * 2048.0f * 256.0f);

// Padded LDS tile layout: 64 codebook rows, row stride 260 floats (260 % 64 = 4
// -> conflict-free-ish ds_load_b64 across the 16 lanes of a half-wave).
static constexpr int kRowF = 260;
static constexpr int kBufF = 64 * kRowF;        // floats per LDS buffer (66560 B)

#if defined(__has_builtin)
#if __has_builtin(__builtin_amdgcn_tensor_load_to_lds) && \
    __has_builtin(__builtin_amdgcn_s_wait_tensorcnt)
#define USE_TDM 1
#endif
#endif
#ifndef USE_TDM
#define USE_TDM 0
#endif

__device__ __forceinline__ v8f wmma_f32(v2f a, v2f b, v8f c) {
    // D = A(16x4 f32) * B(4x16 f32) + C(16x16 f32)  -> v_wmma_f32_16x16x4_f32
    return __builtin_amdgcn_wmma_f32_16x16x4_f32(false, a, false, b,
                                                 (short)0, c, false, false);
}

#if USE_TDM
// TDM: DMA one 64x256 f32 tile (rows n0..n0+63 of E2[b]) into LDS at lds_off,
// padding the LDS destination by 4 DWORDs every 256 DWORDs (row stride 260 f).
// This toolchain (clang-23 flavor) declares the builtin with 6 args:
//   (v4u group0, v8i group1, v4i group2, v4i group3, v8i group4, i32 cpol)
__device__ __forceinline__ void tdm_load_tile(const float* gsrc, unsigned lds_off) {
    unsigned long long ga = (unsigned long long)(uintptr_t)gsrc;
    v4u g0;
    g0.x = 1u;                                        // count=1, user descriptor
    g0.y = lds_off;                                   // lds_addr (bytes)
    g0.z = (unsigned)(ga & 0xFFFFFFFFu);              // global_addr[31:0]
    g0.w = (unsigned)((ga >> 32) & 0x1FFFFFFu)        // global_addr[56:32]
         | (2u << 30);                                // type = 2 ("image")
    v8i g1;
    g1[0] = (int)((2u << 16)      // data_size = 4 bytes
                | (1u << 20)      // pad_enable
                | (7u << 22)      // pad_interval: 256 DWORDs
                | (3u << 25));    // pad_amount:   4 DWORDs
    g1[1] = (int)(((unsigned)kDim & 0xFFFFu) << 16);    // tensor_dim0[15:0] = 256
    g1[2] = (int)(((unsigned)kNEmbed & 0xFFFFu) << 16); // tensor_dim1[15:0] = 512
    g1[3] = (int)(((unsigned)kDim) << 16);              // tile_dim0 = 256
    g1[4] = 64;                                         // tile_dim1 = 64
    g1[5] = kDim;                                       // tensor_dim0_stride = 256
    g1[6] = 0;
    g1[7] = 0;
    v4i z4 = {0, 0, 0, 0};                              // groups 2/3 unused (2D)
    v8i z8 = {0, 0, 0, 0, 0, 0, 0, 0};
    __builtin_amdgcn_tensor_load_to_lds(g0, g1, z4, z4, z8, 0);
}
#endif

// ---------------------------------------------------------------------------
// Stage A: codebook build.  One block per n (512 blocks), 16 waves per block,
// wave w owns d-tile [16w, 16w+16).  cb[n,:,b] = emb[n] @ spk2[b,:], column-
// normalize over d, add spk1; write E2[b][n][d] and S[b][n] = sum_d e^2.
// ---------------------------------------------------------------------------
__global__ __launch_bounds__(512)
void cdna5_codebook(const float* __restrict__ emb,   // (512,256,256)
                    const float* __restrict__ spk,   // (3,16,256)
                    float* __restrict__ E2,          // (16,512,256)
                    float* __restrict__ S)           // (16,512)
{
    const int n    = (int)blockIdx.x;
    const int tid  = (int)threadIdx.x;
    const int wave = tid >> 5;
    const int lane = tid & 31;
    const int half = lane >> 4;
    const int col  = lane & 15;
    const int d0   = wave << 4;

    __shared__ float ssq[16];
    __shared__ float ssq2[16];
    if (tid < 16) { ssq[tid] = 0.0f; ssq2[tid] = 0.0f; }
    __syncthreads();

    const float* Arow = emb + ((size_t)n * kDim + (size_t)(d0 + col)) * kDim;
    const float* Bcol = spk + (size_t)(2 * kB + col) * kDim;

    v8f acc = {};
    for (int k0 = 0; k0 < kDim; k0 += 4) {
        const int k = k0 + 2 * half;
        v2f a = *(const v2f*)(Arow + k);
        v2f b = *(const v2f*)(Bcol + k);
        acc = wmma_f32(a, b, acc);
    }

    float local = 0.0f;
#pragma unroll
    for (int r = 0; r < 8; ++r) local += acc[r] * acc[r];
    atomicAdd(&ssq[col], local);
    __syncthreads();

    const float rn = rsqrtf(ssq[col]);
    const float* addp = spk + (size_t)(kB + col) * kDim + d0 + 8 * half;

    float v[8];
    float local2 = 0.0f;
#pragma unroll
    for (int r = 0; r < 8; ++r) {
        v[r] = acc[r] * rn + addp[r];
        local2 += v[r] * v[r];
    }
    atomicAdd(&ssq2[col], local2);

    float* dst = E2 + ((size_t)col * kNEmbed + n) * kDim + d0 + 8 * half;
    v4f lo = { v[0], v[1], v[2], v[3] };
    v4f hi = { v[4], v[5], v[6], v[7] };
    *(v4f*)(dst)     = lo;
    *(v4f*)(dst + 4) = hi;

    __syncthreads();
    if (tid < 16) S[(size_t)tid * kNEmbed + n] = ssq2[tid];
}

// ---------------------------------------------------------------------------
// Stage B: nearest codeword.  128 blocks (16 b x 8 t-supertiles of 256 rows),
// 8 waves; wave owns two 16-row t-tiles (M=32).  Codebook tiles (64 n-rows)
// are staged in LDS -- TDM double-buffered when available -- and shared by
// all 8 waves.  score(t,n) = S[b][n] - 2 * x.e ; argmin over n.
// ---------------------------------------------------------------------------
__global__ __launch_bounds__(256)
void cdna5_argmin(const float* __restrict__ x,     // (16,2048,256)
                  const float* __restrict__ E2,    // (16,512,256)
                  const float* __restrict__ S,     // (16,512)
                  int* __restrict__ ind,           // (16,2048)
                  float* __restrict__ ind_f)       // (16,2048) float copy
{
    __shared__ float shE[2 * kBufF];

    const int wave = (int)threadIdx.x >> 5;
    const int lane = (int)threadIdx.x & 31;
    const int half = lane >> 4;
    const int col  = lane & 15;
    const int b    = (int)blockIdx.x >> 3;
    const int t0   = ((int)blockIdx.x & 7) * 256 + wave * 32;

    const float* Arow0 = x + ((size_t)b * kT + (size_t)(t0 + col)) * kDim;
    const float* Arow1 = Arow0 + 16 * kDim;
    const float* Eb    = E2 + (size_t)b * kNEmbed * kDim;
    const float* Sb    = S  + (size_t)b * kNEmbed;

    float best0[8], best1[8];
    int   bn0[8],   bn1[8];
#pragma unroll
    for (int r = 0; r < 8; ++r) {
        best0[r] = 3.402823e38f; bn0[r] = 0;
        best1[r] = 3.402823e38f; bn1[r] = 0;
    }

#if USE_TDM
    const unsigned lds_base = (unsigned)(uintptr_t)&shE[0];
    if (threadIdx.x < 32) tdm_load_tile(Eb, lds_base);
#endif

    for (int nb = 0; nb < 8; ++nb) {
        const int n0 = nb * 64;
#if USE_TDM
        if (threadIdx.x < 32) {
            if (nb + 1 < 8) {
                tdm_load_tile(Eb + (size_t)(nb + 1) * 64 * kDim,
                              lds_base + (unsigned)(((nb + 1) & 1) * kBufF * 4));
                __builtin_amdgcn_s_wait_tensorcnt(1);   // tile nb landed
            } else {
                __builtin_amdgcn_s_wait_tensorcnt(0);
            }
        }
        __syncthreads();
        const int sbase = (nb & 1) * kBufF;
#else
        for (int i = (int)threadIdx.x; i < 64 * 64; i += 256) {
            const int r = i >> 6;
            const int c = (i & 63) << 2;
            v4f tmp = *(const v4f*)(Eb + (size_t)(n0 + r) * kDim + c);
            *(v4f*)&shE[r * kRowF + c] = tmp;
        }
        __syncthreads();
        const int sbase = 0;
#endif
        v8f a00 = {}, a01 = {}, a02 = {}, a03 = {};
        v8f a10 = {}, a11 = {}, a12 = {}, a13 = {};
        for (int k0 = 0; k0 < kDim; k0 += 4) {
            const int k = k0 + 2 * half;
            v2f av0 = *(const v2f*)(Arow0 + k);
            v2f av1 = *(const v2f*)(Arow1 + k);
            v2f bb0 = *(const v2f*)&shE[sbase + (col)      * kRowF + k];
            v2f bb1 = *(const v2f*)&shE[sbase + (16 + col) * kRowF + k];
            v2f bb2 = *(const v2f*)&shE[sbase + (32 + col) * kRowF + k];
            v2f bb3 = *(const v2f*)&shE[sbase + (48 + col) * kRowF + k];
            a00 = wmma_f32(av0, bb0, a00);
            a10 = wmma_f32(av1, bb0, a10);
            a01 = wmma_f32(av0, bb1, a01);
            a11 = wmma_f32(av1, bb1, a11);
            a02 = wmma_f32(av0, bb2, a02);
            a12 = wmma_f32(av1, bb2, a12);
            a03 = wmma_f32(av0, bb3, a03);
            a13 = wmma_f32(av1, bb3, a13);
        }
        const float s0 = Sb[n0 + col];
        const float s1 = Sb[n0 + 16 + col];
        const float s2 = Sb[n0 + 32 + col];
        const float s3 = Sb[n0 + 48 + col];
#pragma unroll
        for (int r = 0; r < 8; ++r) {
            float sc;
            sc = s0 - 2.0f * a00[r]; if (sc < best0[r]) { best0[r] = sc; bn0[r] = n0 + col; }
            sc = s1 - 2.0f * a01[r]; if (sc < best0[r]) { best0[r] = sc; bn0[r] = n0 + 16 + col; }
            sc = s2 - 2.0f * a02[r]; if (sc < best0[r]) { best0[r] = sc; bn0[r] = n0 + 32 + col; }
            sc = s3 - 2.0f * a03[r]; if (sc < best0[r]) { best0[r] = sc; bn0[r] = n0 + 48 + col; }
            sc = s0 - 2.0f * a10[r]; if (sc < best1[r]) { best1[r] = sc; bn1[r] = n0 + col; }
            sc = s1 - 2.0f * a11[r]; if (sc < best1[r]) { best1[r] = sc; bn1[r] = n0 + 16 + col; }
            sc = s2 - 2.0f * a12[r]; if (sc < best1[r]) { best1[r] = sc; bn1[r] = n0 + 32 + col; }
            sc = s3 - 2.0f * a13[r]; if (sc < best1[r]) { best1[r] = sc; bn1[r] = n0 + 48 + col; }
        }
        __syncthreads();   // done reading this LDS buffer before it is reloaded
    }

    // 16-lane argmin reduction; tie-break toward smaller n (argmax first-hit).
#pragma unroll
    for (int off = 1; off < 16; off <<= 1) {
#pragma unroll
        for (int r = 0; r < 8; ++r) {
            float ov; int oi;
            ov = __shfl_xor(best0[r], off, 32);
            oi = __shfl_xor(bn0[r],   off, 32);
            if (ov < best0[r] || (ov == best0[r] && oi < bn0[r])) { best0[r] = ov; bn0[r] = oi; }
            ov = __shfl_xor(best1[r], off, 32);
            oi = __shfl_xor(bn1[r],   off, 32);
            if (ov < best1[r] || (ov == best1[r] && oi < bn1[r])) { best1[r] = ov; bn1[r] = oi; }
        }
    }

    if (col == 0) {
#pragma unroll
        for (int r = 0; r < 8; ++r) {
            const int t = t0 + r + 8 * half;
            ind[(size_t)b * kT + t]        = bn0[r];
            ind_f[(size_t)b * kT + t]      = (float)bn0[r];
            ind[(size_t)b * kT + t + 16]   = bn1[r];
            ind_f[(size_t)b * kT + t + 16] = (float)bn1[r];
        }
    }
}

// ---------------------------------------------------------------------------
// Stage C: gather quantize = E2[b][ind][*] (coalesced in d) + MSE reduction.
// ---------------------------------------------------------------------------
__global__ __launch_bounds__(256)
void cdna5_gather(const float* __restrict__ x,
                  const float* __restrict__ E2,
                  const int* __restrict__ ind,
                  float* __restrict__ outq,
                  float* __restrict__ diff)
{
    const int row = (int)blockIdx.x;           // b*2048 + t
    const int d   = (int)threadIdx.x;
    const int b   = row >> 11;
    const int n   = ind[row];

    const float q  = E2[((size_t)b * kNEmbed + n) * kDim + d];
    const float xv = x[(size_t)row * kDim + d];
    outq[(size_t)row * kDim + d] = q;          // (quantize + quantize_1)/2 == quantize

    float p = (q - xv) * (q - xv);

    __shared__ float red[8];
    for (int off = 16; off; off >>= 1) p += __shfl_down(p, off, 32);
    if ((threadIdx.x & 31) == 0) red[threadIdx.x >> 5] = p;
    __syncthreads();
    if (threadIdx.x < 32) {
        float v = (threadIdx.x < 8) ? red[threadIdx.x] : 0.0f;
        for (int off = 16; off; off >>= 1) v += __shfl_down(v, off, 32);
        if (threadIdx.x == 0) atomicAdd(diff, v * kInvTotal);
    }
}

__global__ void cdna5_zero(float* __restrict__ p) { *p = 0.0f; }

// ---------------------------------------------------------------------------
extern "C" void kernel_launch(void* const* d_in, const int* in_sizes, int n_in,
                              void* d_out, int out_size, void* d_ws, size_t ws_size,
                              hipStream_t stream)
{
    const float* x   = (const float*)d_in[0];   // (16,2048,256)
    const float* spk = (const float*)d_in[1];   // (3,16,256)
    const float* emb = (const float*)d_in[2];   // (512,256,256)

    char* ws = (char*)d_ws;
    float* E2 = (float*)ws;                                           // 8 MiB
    float* S  = (float*)(ws + (size_t)kB * kNEmbed * kDim * sizeof(float));
    int*   ind = (int*)(ws + (size_t)kB * kNEmbed * kDim * sizeof(float)
                           + (size_t)kB * kNEmbed * sizeof(float));

    float* outq  = (float*)d_out;
    float* diff  = outq + (size_t)kB * kT * kDim;   // element 8388608
    float* ind_f = diff + 1;                        // 32768 floats

    cdna5_zero    <<<1, 1, 0, stream>>>(diff);
    cdna5_codebook<<<kNEmbed, 512, 0, stream>>>(emb, spk, E2, S);
    cdna5_argmin  <<<kB * 8, 256, 0, stream>>>(x, E2, S, ind, ind_f);
    cdna5_gather  <<<kB * kT, 256, 0, stream>>>(x, E2, ind, outq, diff);
}